// AttPoolBlock_14620068675983
// MI455X (gfx1250) — compile-verified
//
#include <hip/hip_runtime.h>

typedef float v2f __attribute__((ext_vector_type(2)));
typedef float v8f __attribute__((ext_vector_type(8)));

#define B_   8
#define N_   2048
#define D_   128
#define C_   204
#define KMAX 512

static __device__ __forceinline__ v8f zero8() {
  v8f v;
#pragma unroll
  for (int i = 0; i < 8; ++i) v[i] = 0.0f;
  return v;
}

// ---------------- Kernel 1: ref_hidden = relu((sum_n X) @ w) ----------------
__global__ __launch_bounds__(128) void k_refhidden(const float* __restrict__ X,
                                                   const float* __restrict__ w,
                                                   float* __restrict__ refh) {
  int b = blockIdx.x;
  int d = threadIdx.x;
  __shared__ float hid[D_];
  const float* Xb = X + (size_t)b * N_ * D_;
  float s = 0.0f;
  for (int n = 0; n < N_; ++n) s += Xb[(size_t)n * D_ + d];
  hid[d] = s;
  __syncthreads();
  float acc = 0.0f;
  for (int j = 0; j < D_; ++j) acc += hid[j] * w[(size_t)j * D_ + d];
  refh[b * D_ + d] = fmaxf(acc, 0.0f);
}

// ---------------- Kernel 2: inner_prod[b,n] = X[b,n,:] . refh[b,:] ----------
__global__ __launch_bounds__(256) void k_scores(const float* __restrict__ X,
                                                const float* __restrict__ refh,
                                                float* __restrict__ scores) {
  int gid  = blockIdx.x * (blockDim.x >> 5) + (threadIdx.x >> 5); // global wave id
  int lane = threadIdx.x & 31;
  int b = gid >> 11;          // /2048
  int n = gid & (N_ - 1);
  const float* x = X + ((size_t)b * N_ + n) * D_;
  const float* r = refh + b * D_;
  float s = 0.0f;
#pragma unroll
  for (int i = 0; i < D_ / 32; ++i) s += x[lane + 32 * i] * r[lane + 32 * i];
#pragma unroll
  for (int off = 16; off > 0; off >>= 1) s += __shfl_down(s, off, 32);
  if (lane == 0) scores[gid] = s;
}

// ---------------- Kernel 3: exact stable top-512 via rank counting ----------
__global__ __launch_bounds__(512) void k_topk(const float* __restrict__ scores,
                                              int* __restrict__ top_index) {
  int b = blockIdx.x;
  __shared__ float v[N_];
  for (int i = threadIdx.x; i < N_; i += blockDim.x) v[i] = scores[b * N_ + i];
  __syncthreads();
  for (int i = threadIdx.x; i < N_; i += blockDim.x) {
    float vi = v[i];
    int rank = 0;
    for (int j = 0; j < N_; ++j) {
      float vj = v[j];
      rank += (vj > vi) || (vj == vi && j < i);
    }
    if (rank < KMAX) top_index[b * KMAX + rank] = i;
  }
}

// ---------------- Kernel 4: k_list = ceil(0.25 * count(mask)) ---------------
__global__ __launch_bounds__(256) void k_klist(const float* __restrict__ mask,
                                               int* __restrict__ klist) {
  int b = blockIdx.x;
  __shared__ int wsum[8];
  int lane = threadIdx.x & 31, wv = threadIdx.x >> 5;
  int c = 0;
  for (int i = threadIdx.x; i < N_; i += 256) c += (mask[b * N_ + i] != 0.0f) ? 1 : 0;
#pragma unroll
  for (int off = 16; off > 0; off >>= 1) c += __shfl_down(c, off, 32);
  if (lane == 0) wsum[wv] = c;
  __syncthreads();
  if (threadIdx.x == 0) {
    int t = 0;
#pragma unroll
    for (int i = 0; i < 8; ++i) t += wsum[i];
    klist[b] = (t + 3) >> 2;   // ceil(t/4)
  }
}

// ---------------- Kernel 5: fused gathered GEMM + InterChannelGCN -----------
// Per block: batch b, 64-row M tile. 256 threads = 8 waves.
// P[64,128] = adj[gather,:]@X + assign[gather,:]@Hc ; out = relu(P @ w_ic)
//
// A tile: row-major [64][ASTRIDE]  -> A frag = one aligned 8B LDS read
// B tile: k-pair interleaved: (k,n) at (k>>1)*PSTRIDE + 2n + (k&1)
//         -> B frag = one aligned 8B LDS read (no packing movs), and
//            fills hit all 64 LDS banks exactly once.
#define ASTRIDE 68
#define PSTRIDE 264            // 2*128 + 8 pad floats per k-pair row
#define TSTRIDE 132            // P tile row stride when parked in Bt
// Bt floats: 32*PSTRIDE = 8448 == 64*TSTRIDE (P tile reuses Bt exactly)

__global__ __launch_bounds__(256) void k_pool_gemm(
    const float* __restrict__ X, const float* __restrict__ adj,
    const float* __restrict__ assign, const float* __restrict__ Hc,
    const float* __restrict__ w_ic, const int* __restrict__ top_index,
    const int* __restrict__ klist, float* __restrict__ Hout) {
  int b  = blockIdx.y;
  int m0 = blockIdx.x * 64;
  __shared__ float At[64 * ASTRIDE];
  __shared__ float Bt[32 * PSTRIDE];
  __shared__ int sidx[64];

  int tid  = threadIdx.x;
  int lane = tid & 31;
  int wv   = tid >> 5;      // 0..7
  int mw   = wv & 3;        // which 16-row M tile
  int nw   = wv >> 2;       // which 64-col N half
  int mo   = mw * 16;
  int kb   = klist[b];

  if (tid < 64) {
    int m = m0 + tid;
    sidx[tid] = (m < kb) ? top_index[b * KMAX + m] : -1;  // -1 => zero row
  }
  __syncthreads();

  const float* adjb = adj    + (size_t)b * N_ * N_;
  const float* Xb   = X      + (size_t)b * N_ * D_;
  const float* asb  = assign + (size_t)b * N_ * C_;
  const float* Hcb  = Hc     + (size_t)b * C_ * D_;

  // per-lane WMMA fragment coordinates (ISA 7.12.2, 32-bit A 16x4 / C,D 16x16)
  int fm = lane & 15;             // M (A) or N (B) within tile
  int fh = lane >> 4;             // 0: K+0,K+1 ; 1: K+2,K+3
  int fk = fh << 1;

  v8f acc[4];
#pragma unroll
  for (int t = 0; t < 4; ++t) acc[t] = zero8();

  // ---- Stage 1a: gathered adj rows @ X, K = 2048 in chunks of 64 ----
  for (int k0 = 0; k0 < N_; k0 += 64) {
    for (int i = tid; i < 64 * 64; i += 256) {
      int r = i >> 6, c = i & 63;
      int g = sidx[r];
      At[r * ASTRIDE + c] = (g >= 0) ? adjb[(size_t)g * N_ + k0 + c] : 0.0f;
    }
    // X tile in k-pair interleaved layout: one b64 store per thread-iter
    for (int i = tid; i < 32 * 128; i += 256) {
      int p = i >> 7, c = i & 127;
      float2 x2;
      x2.x = Xb[(size_t)(k0 + 2 * p)     * D_ + c];
      x2.y = Xb[(size_t)(k0 + 2 * p + 1) * D_ + c];
      *(float2*)&Bt[p * PSTRIDE + 2 * c] = x2;
    }
    __syncthreads();
#pragma unroll
    for (int kk = 0; kk < 16; ++kk) {
      int kbase = kk * 4 + fk;
      v2f a = *(const v2f*)&At[(mo + fm) * ASTRIDE + kbase];
      int prow = kk * 2 + fh;
#pragma unroll
      for (int t = 0; t < 4; ++t) {
        int nc = nw * 64 + t * 16;
        v2f bf = *(const v2f*)&Bt[prow * PSTRIDE + 2 * (nc + fm)];
        acc[t] = __builtin_amdgcn_wmma_f32_16x16x4_f32(
            false, a, false, bf, (short)0, acc[t], false, false);
      }
    }
    __syncthreads();
  }

  // ---- Stage 1b: gathered assign rows @ H_coarse, C = 204 (pad to 256) ----
  for (int c0 = 0; c0 < 256; c0 += 64) {
    for (int i = tid; i < 64 * 64; i += 256) {
      int r = i >> 6, c = i & 63;
      int g = sidx[r];
      At[r * ASTRIDE + c] =
          (g >= 0 && (c0 + c) < C_) ? asb[(size_t)g * C_ + c0 + c] : 0.0f;
    }
    for (int i = tid; i < 32 * 128; i += 256) {
      int p = i >> 7, c = i & 127;
      int r0 = c0 + 2 * p, r1 = r0 + 1;
      float2 x2;
      x2.x = (r0 < C_) ? Hcb[(size_t)r0 * D_ + c] : 0.0f;
      x2.y = (r1 < C_) ? Hcb[(size_t)r1 * D_ + c] : 0.0f;
      *(float2*)&Bt[p * PSTRIDE + 2 * c] = x2;
    }
    __syncthreads();
#pragma unroll
    for (int kk = 0; kk < 16; ++kk) {
      int kbase = kk * 4 + fk;
      v2f a = *(const v2f*)&At[(mo + fm) * ASTRIDE + kbase];
      int prow = kk * 2 + fh;
#pragma unroll
      for (int t = 0; t < 4; ++t) {
        int nc = nw * 64 + t * 16;
        v2f bf = *(const v2f*)&Bt[prow * PSTRIDE + 2 * (nc + fm)];
        acc[t] = __builtin_amdgcn_wmma_f32_16x16x4_f32(
            false, a, false, bf, (short)0, acc[t], false, false);
      }
    }
    __syncthreads();
  }

  // ---- Park P tile (64x128, row-major) in Bt ----
  {
    int n  = lane & 15;
    int mr = fh * 8;
#pragma unroll
    for (int t = 0; t < 4; ++t) {
      int nc = nw * 64 + t * 16;
#pragma unroll
      for (int r = 0; r < 8; ++r)
        Bt[(mo + mr + r) * TSTRIDE + nc + n] = acc[t][r];
    }
  }
  __syncthreads();

  // ---- Stage 2: out = relu(P @ w_ic), K = 128 ----
  v8f o[4];
#pragma unroll
  for (int t = 0; t < 4; ++t) o[t] = zero8();
#pragma unroll 4
  for (int kk = 0; kk < 32; ++kk) {
    int kbase = kk * 4 + fk;
    v2f a = *(const v2f*)&Bt[(mo + fm) * TSTRIDE + kbase];
#pragma unroll
    for (int t = 0; t < 4; ++t) {
      int nc = nw * 64 + t * 16;
      v2f bf;
      bf.x = w_ic[(size_t)(kbase)     * D_ + nc + fm];
      bf.y = w_ic[(size_t)(kbase + 1) * D_ + nc + fm];
      o[t] = __builtin_amdgcn_wmma_f32_16x16x4_f32(
          false, a, false, bf, (short)0, o[t], false, false);
    }
  }

  // ---- Store with relu ----
  {
    int n  = lane & 15;
    int mr = fh * 8;
    float* Hb = Hout + ((size_t)b * KMAX + m0) * D_;
#pragma unroll
    for (int t = 0; t < 4; ++t) {
      int nc = nw * 64 + t * 16;
#pragma unroll
      for (int r = 0; r < 8; ++r)
        Hb[(size_t)(mo + mr + r) * D_ + nc + n] = fmaxf(o[t][r], 0.0f);
    }
  }
}

// ---------------------------------------------------------------------------
extern "C" void kernel_launch(void* const* d_in, const int* in_sizes, int n_in,
                              void* d_out, int out_size, void* d_ws, size_t ws_size,
                              hipStream_t stream) {
  const float* X    = (const float*)d_in[0];   // [8,2048,128]
  const float* adj  = (const float*)d_in[1];   // [8,2048,2048]
  const float* mask = (const float*)d_in[2];   // [8,2048]
  const float* asn  = (const float*)d_in[3];   // [8,2048,204]
  const float* Hc   = (const float*)d_in[4];   // [8,204,128]
  const float* w    = (const float*)d_in[5];   // [128,128]
  const float* w_ic = (const float*)d_in[6];   // [128,128]

  // d_out layout (tuple flattened): top_index[8*512] i32, H[8*512*128] f32, k_list[8] i32
  char* out = (char*)d_out;
  int*   top_index = (int*)out;
  float* Hout      = (float*)(out + (size_t)B_ * KMAX * 4);
  int*   klist     = (int*)(out + (size_t)B_ * KMAX * 4 + (size_t)B_ * KMAX * D_ * 4);

  // workspace: ref_hidden [8*128] f32, then scores [8*2048] f32
  float* refh   = (float*)d_ws;
  float* scores = (float*)((char*)d_ws + (size_t)B_ * D_ * 4);

  k_refhidden<<<B_, D_, 0, stream>>>(X, w, refh);
  k_scores<<<(B_ * N_) / 8, 256, 0, stream>>>(X, refh, scores);   // 8 waves/block
  k_topk<<<B_, 512, 0, stream>>>(scores, top_index);
  k_klist<<<B_, 256, 0, stream>>>(mask, klist);
  dim3 grid(KMAX / 64, B_);     // 8 M-tiles x 8 batches
  k_pool_gemm<<<grid, 256, 0, stream>>>(X, adj, asn, Hc, w_ic, top_index, klist, Hout);
}